// NodeModel_43722767073862
// MI455X (gfx1250) — compile-verified
//
#include <hip/hip_runtime.h>
#include <math.h>

typedef __attribute__((ext_vector_type(2))) float v2f;
typedef __attribute__((ext_vector_type(8))) float v8f;

#define DCH 128
#define LDS_S 130   // padded LDS stride: even (b64 alignment), 130%64=2 spreads banks

// ---------------- degree / normalization ----------------

__global__ void deg_init_kernel(float* __restrict__ deg, int n) {
    int i = blockIdx.x * blockDim.x + threadIdx.x;
    if (i < n) deg[i] = 1.0f;   // self-loop contributes 1
}

__global__ void deg_accum_kernel(const long long* __restrict__ dst,
                                 float* __restrict__ deg, long long nE) {
    long long e = (long long)blockIdx.x * blockDim.x + threadIdx.x;
    if (e < nE) atomicAdd(&deg[dst[e]], 1.0f);
}

__global__ void deg_rsqrt_kernel(float* __restrict__ deg, int n) {
    int i = blockIdx.x * blockDim.x + threadIdx.x;
    if (i < n) {
        float v = deg[i];
        deg[i] = (v > 0.0f) ? rsqrtf(v) : 0.0f;   // becomes dis[]
    }
}

// ---------------- fp32 WMMA GEMM: H = X @ W  (X: nrows x 128, W: 128 x 128) ----------------
// Block = 256 threads = 8 wave32 waves; each wave owns one 16-row tile across
// all 8 col tiles (64 VGPRs of f32 accumulators). W is staged TRANSPOSED in
// LDS with a padded stride so each 4x16 B fragment is one aligned ds_load_b64
// (K-adjacent pair contiguous) -> no repacking moves between WMMAs.

__global__ __launch_bounds__(256) void gemm_wmma_kernel(
    const float* __restrict__ X, const float* __restrict__ W,
    float* __restrict__ H, int nrows) {
    __shared__ float Wt[DCH * LDS_S];   // ~65 KB of the 320 KB WGP LDS

    int tid = threadIdx.x;
    // cooperative transposed load: read W row-major (coalesced float4),
    // scatter into Wt[col*LDS_S + k]
    for (int i = tid * 4; i < DCH * DCH; i += 256 * 4) {
        int r = i >> 7;        // k row
        int c = i & 127;       // col
        float4 w = *(const float4*)&W[i];
        Wt[(c + 0) * LDS_S + r] = w.x;
        Wt[(c + 1) * LDS_S + r] = w.y;
        Wt[(c + 2) * LDS_S + r] = w.z;
        Wt[(c + 3) * LDS_S + r] = w.w;
    }
    __syncthreads();

    int wave = tid >> 5;
    int lane = tid & 31;
    int rowBase = (blockIdx.x * 8 + wave) * 16;   // 16-row tile per wave
    if (rowBase >= nrows) return;                 // whole-wave early out (EXEC stays full)

    int half = lane >> 4;    // 0: lanes 0-15, 1: lanes 16-31
    int l16  = lane & 15;
    int arow = rowBase + l16;

    v8f acc[8] = {};         // 8 col tiles x (16x16 f32 / 32 lanes)

    for (int k0 = 0; k0 < DCH; k0 += 4) {
        // A fragment (16x4 f32): lane<16 holds K=k0,k0+1; lane>=16 holds K=k0+2,k0+3
        int kb = k0 + half * 2;
        v2f a = *(const v2f*)&X[(size_t)arow * DCH + kb];
#pragma unroll
        for (int n = 0; n < 8; ++n) {
            // B fragment (4x16 f32): one aligned 64-bit LDS load per fragment
            int col = n * 16 + l16;
            v2f b = *(const v2f*)&Wt[col * LDS_S + kb];
            acc[n] = __builtin_amdgcn_wmma_f32_16x16x4_f32(
                false, a, false, b, (short)0, acc[n], false, false);
        }
    }

    // C/D layout: VGPR r -> row rowBase + r + half*8, col = n*16 + l16
#pragma unroll
    for (int n = 0; n < 8; ++n) {
        int col = n * 16 + l16;
#pragma unroll
        for (int r = 0; r < 8; ++r) {
            int row = rowBase + r + half * 8;
            H[(size_t)row * DCH + col] = acc[n][r];
        }
    }
}

// ---------------- out = bias + h * dis^2  (self-loop folded in) ----------------

__global__ void init_out_kernel(const float* __restrict__ H,
                                const float* __restrict__ bias,
                                const float* __restrict__ dis,
                                float* __restrict__ out, int n) {
    int i = blockIdx.x * blockDim.x + threadIdx.x;   // float4 index
    if (i >= n * 32) return;
    int node = i >> 5;
    int c = (i & 31) << 2;
    float s = dis[node]; s = s * s;
    float4 h = *(const float4*)&H[(size_t)node * DCH + c];
    float4 b = *(const float4*)&bias[c];
    float4 o;
    o.x = b.x + h.x * s;  o.y = b.y + h.y * s;
    o.z = b.z + h.z * s;  o.w = b.w + h.w * s;
    *(float4*)&out[(size_t)node * DCH + c] = o;
}

// ---------------- edge scatter: out[dst] += h[src] * dis[src]*dis[dst] ----------------
// 32 threads per edge, float4 per thread. h/out (25.6 MB each) both live in
// the 192 MB L2, so random gathers and the f32 atomics resolve on-chip.

__global__ void scatter_kernel(const float* __restrict__ H,
                               const long long* __restrict__ src,
                               const long long* __restrict__ dst,
                               const float* __restrict__ dis,
                               float* __restrict__ out, long long nE) {
    long long t = (long long)blockIdx.x * blockDim.x + threadIdx.x;
    long long e = t >> 5;
    if (e >= nE) return;
    int c = (int)(t & 31) << 2;
    long long s = src[e];
    long long d = dst[e];
    float norm = dis[s] * dis[d];
    float4 h = *(const float4*)&H[s * DCH + c];
    float* o = &out[d * DCH + c];
    atomicAdd(o + 0, h.x * norm);
    atomicAdd(o + 1, h.y * norm);
    atomicAdd(o + 2, h.z * norm);
    atomicAdd(o + 3, h.w * norm);
}

// ---------------- elementwise ----------------

__global__ void relu_kernel(float* __restrict__ x, int n4) {
    int i = blockIdx.x * blockDim.x + threadIdx.x;
    if (i >= n4) return;
    float4 v = *(float4*)&x[(size_t)i * 4];
    v.x = fmaxf(v.x, 0.f); v.y = fmaxf(v.y, 0.f);
    v.z = fmaxf(v.z, 0.f); v.w = fmaxf(v.w, 0.f);
    *(float4*)&x[(size_t)i * 4] = v;
}

// mu = mu + eps * exp(logstd) / 10  (in place on mu)
__global__ void reparam_kernel(float* __restrict__ mu,
                               const float* __restrict__ eps,
                               const float* __restrict__ ls, int n4) {
    int i = blockIdx.x * blockDim.x + threadIdx.x;
    if (i >= n4) return;
    size_t o = (size_t)i * 4;
    float4 m = *(float4*)&mu[o];
    float4 e = *(const float4*)&eps[o];
    float4 l = *(const float4*)&ls[o];
    m.x += e.x * __expf(l.x) * 0.1f;
    m.y += e.y * __expf(l.y) * 0.1f;
    m.z += e.z * __expf(l.z) * 0.1f;
    m.w += e.w * __expf(l.w) * 0.1f;
    *(float4*)&mu[o] = m;
}

// ---------------- host orchestration ----------------

extern "C" void kernel_launch(void* const* d_in, const int* in_sizes, int n_in,
                              void* d_out, int out_size, void* d_ws, size_t ws_size,
                              hipStream_t stream) {
    const float*     x    = (const float*)d_in[0];
    const long long* ei   = (const long long*)d_in[1];   // int64 per reference
    const float*     eps  = (const float*)d_in[2];
    const float* W1 = (const float*)d_in[3],  *b1 = (const float*)d_in[4];
    const float* W2 = (const float*)d_in[5],  *b2 = (const float*)d_in[6];
    const float* Wm = (const float*)d_in[7],  *bm = (const float*)d_in[8];
    const float* Wl = (const float*)d_in[9],  *bl = (const float*)d_in[10];
    const float* W5 = (const float*)d_in[11], *b5 = (const float*)d_in[12];
    const float* W6 = (const float*)d_in[13], *b6 = (const float*)d_in[14];

    const int  ND = in_sizes[0];
    const int  N  = ND / DCH;
    const long long E = in_sizes[1] / 2;
    const long long* srcI = ei;
    const long long* dstI = ei + E;

    float* ws  = (float*)d_ws;
    float* dis = ws;                       // N floats (deg -> dis in place)
    float* H   = dis + N;                  // N*D GEMM scratch
    float* B0  = H  + (size_t)N * DCH;     // N*D
    float* B1  = B0 + (size_t)N * DCH;     // N*D
    float* LS  = B1 + (size_t)N * DCH;     // N*D (logstd)

    const int T = 256;
    const int gN   = (N + T - 1) / T;
    const int gE   = (int)((E + T - 1) / T);
    const int n4   = ND / 4;
    const int gND4 = (n4 + T - 1) / T;
    const int gGEMM = ((N + 15) / 16 + 7) / 8;
    const int gSCAT = (int)((E * 32 + T - 1) / T);

    // normalization: deg = 1 + scatter(ones over dst); dis = rsqrt(deg)
    deg_init_kernel<<<gN, T, 0, stream>>>(dis, N);
    deg_accum_kernel<<<gE, T, 0, stream>>>(dstI, dis, E);
    deg_rsqrt_kernel<<<gN, T, 0, stream>>>(dis, N);

    auto layer = [&](const float* in, const float* W, const float* b,
                     float* out, bool relu) {
        gemm_wmma_kernel<<<gGEMM, T, 0, stream>>>(in, W, H, N);
        init_out_kernel<<<gND4, T, 0, stream>>>(H, b, dis, out, N);
        scatter_kernel<<<gSCAT, T, 0, stream>>>(H, srcI, dstI, dis, out, E);
        if (relu) relu_kernel<<<gND4, T, 0, stream>>>(out, n4);
    };

    float* OUT = (float*)d_out;

    layer(x,  W1, b1, B0, true);    // nodes1 = relu(gcn(x))
    layer(B0, W2, b2, B1, true);    // nodes2 = relu(gcn(nodes1))
    layer(B1, Wm, bm, B0, false);   // mu      -> B0
    layer(B1, Wl, bl, LS, false);   // logstd  -> LS
    reparam_kernel<<<gND4, T, 0, stream>>>(B0, eps, LS, n4);  // B0 = mu + eps*exp(ls)/10
    layer(B0, W5, b5, B1, true);    // nodes5
    layer(B1, W6, b6, OUT, true);   // final -> d_out
}